// Attention_65644280152585
// MI455X (gfx1250) — compile-verified
//
#include <hip/hip_runtime.h>
#include <math.h>

typedef __attribute__((ext_vector_type(2))) float v2f;
typedef __attribute__((ext_vector_type(8))) float v8f;

#define N_TOK 196
#define C_CH  480
#define M_KV  23520   // (C/4)*N
#define I_EPS 1e-3f

// ---------------------------------------------------------------------------
// ws layout (float offsets)
#define WS_STATS 0        // [0]=sumA [1]=sumA2, [2+2h]/[3+2h]=K stats, [16+2i]/[17+2i]=Q stats (i=b*4+h)
#define WS_SCL   64       // 16 per-(branch,head) softmax scales
#define WS_QC    80
#define WS_KC    (WS_QC   + N_TOK*C_CH)
#define WS_VC    (WS_KC   + N_TOK*C_CH)
#define WS_ATTN  (WS_VC   + N_TOK*C_CH)     // 480*480, softmax done in place
#define WS_KVS   (WS_ATTN + C_CH*C_CH)      // 23520 x 4
#define WS_KH    (WS_KVS  + M_KV*4)         // 4 x 23520
#define WS_VH    (WS_KH   + M_KV*4)
#define WS_QARR  (WS_VH   + M_KV*4)         // 16 x 196
#define WS_CARR  (WS_QARR + 16*N_TOK)       // 16 x 196

#define WMMA4(a, b, c) \
  __builtin_amdgcn_wmma_f32_16x16x4_f32(false, (a), false, (b), (short)0, (c), false, false)

// ---------------------------------------------------------------------------
__global__ void init_stats(float* __restrict__ stats) {
  if (threadIdx.x < 64) stats[threadIdx.x] = 0.0f;
}

// ---- Qc/Kc/Vc = emb_C @ W   (one wave per 16x32 tile, WMMA f32 16x16x4) ----
// A rows are independent in D = A*B, so out-of-range token rows are clamped
// (never stored) -> branchless inner loop, no exec-mask churn.
__global__ __launch_bounds__(32)
void qkv_gemm(const float* __restrict__ emb,
              const float* __restrict__ WqC, const float* __restrict__ WkC,
              const float* __restrict__ WvC,
              float* __restrict__ Qc, float* __restrict__ Kc, float* __restrict__ Vc)
{
  const float* W   = blockIdx.z == 0 ? WqC : (blockIdx.z == 1 ? WkC : WvC);
  float*       Out = blockIdx.z == 0 ? Qc  : (blockIdx.z == 1 ? Kc  : Vc);
  const int m0 = blockIdx.x * 16, n0 = blockIdx.y * 32;
  const int lane = threadIdx.x & 31, half = lane >> 4, l = lane & 15;
  const int m  = m0 + l;
  const int mA = m < N_TOK ? m : (N_TOK - 1);         // clamp: rows >=196 are never stored
  const float* arow = emb + mA * C_CH;
  v8f acc0 = {}, acc1 = {};
  for (int k0 = 0; k0 < C_CH; k0 += 4) {
    const int ka = k0 + 2 * half;                     // even -> 8B-aligned v2f load
    const v2f a = *(const v2f*)(arow + ka);
    v2f b0, b1;
    b0.x = W[ka * C_CH + n0 + l];       b0.y = W[(ka + 1) * C_CH + n0 + l];
    b1.x = W[ka * C_CH + n0 + 16 + l];  b1.y = W[(ka + 1) * C_CH + n0 + 16 + l];
    acc0 = WMMA4(a, b0, acc0);
    acc1 = WMMA4(a, b1, acc1);
  }
  for (int i = 0; i < 8; ++i) {
    const int mm = m0 + half * 8 + i;
    if (mm < N_TOK) {
      Out[mm * C_CH + n0 + l]      = acc0[i];
      Out[mm * C_CH + n0 + 16 + l] = acc1[i];
    }
  }
}

// ---- attn = Qc^T @ Kc (480x480, K=196) + global sum/sumsq for psi1 ----
__global__ __launch_bounds__(32)
void attn_gemm(const float* __restrict__ Qc, const float* __restrict__ Kc,
               float* __restrict__ attn, float* __restrict__ stats)
{
  const int m0 = blockIdx.x * 16, n0 = blockIdx.y * 32;
  const int lane = threadIdx.x & 31, half = lane >> 4, l = lane & 15;
  v8f acc0 = {}, acc1 = {};
  for (int k0 = 0; k0 < N_TOK; k0 += 4) {             // 196 = 49 * 4
    const int ka = k0 + 2 * half;
    v2f a, b0, b1;
    a.x  = Qc[ka * C_CH + m0 + l];      a.y  = Qc[(ka + 1) * C_CH + m0 + l];
    b0.x = Kc[ka * C_CH + n0 + l];      b0.y = Kc[(ka + 1) * C_CH + n0 + l];
    b1.x = Kc[ka * C_CH + n0 + 16 + l]; b1.y = Kc[(ka + 1) * C_CH + n0 + 16 + l];
    acc0 = WMMA4(a, b0, acc0);
    acc1 = WMMA4(a, b1, acc1);
  }
  float s = 0.0f, s2 = 0.0f;
  for (int i = 0; i < 8; ++i) {
    const int mm = m0 + half * 8 + i;
    attn[mm * C_CH + n0 + l]      = acc0[i];
    attn[mm * C_CH + n0 + 16 + l] = acc1[i];
    s += acc0[i] + acc1[i];
    s2 += acc0[i] * acc0[i] + acc1[i] * acc1[i];
  }
  atomicAdd(&stats[0], s);
  atomicAdd(&stats[1], s2);
}

// ---- psi1 (whole-matrix instance norm) + row softmax, in place ----
__global__ __launch_bounds__(256)
void psi1_softmax(float* __restrict__ attn, const float* __restrict__ stats,
                  const float* __restrict__ g1, const float* __restrict__ b1)
{
  __shared__ float red[256];
  const int row = blockIdx.x, t = threadIdx.x;
  const float inv = 1.0f / ((float)C_CH * (float)C_CH);
  const float mean = stats[0] * inv;
  const float var  = stats[1] * inv - mean * mean;
  const float rs   = rsqrtf(var + I_EPS);
  const float g = g1[0], bb = b1[0];
  const bool  has2 = (t + 256) < C_CH;
  float v0 = (attn[row * C_CH + t] - mean) * rs * g + bb;
  float v1 = has2 ? (attn[row * C_CH + t + 256] - mean) * rs * g + bb : -3.0e38f;
  red[t] = fmaxf(v0, v1); __syncthreads();
  for (int off = 128; off; off >>= 1) { if (t < off) red[t] = fmaxf(red[t], red[t + off]); __syncthreads(); }
  const float mx = red[0]; __syncthreads();
  const float e0 = __expf(v0 - mx);
  const float e1 = has2 ? __expf(v1 - mx) : 0.0f;
  red[t] = e0 + e1; __syncthreads();
  for (int off = 128; off; off >>= 1) { if (t < off) red[t] += red[t + off]; __syncthreads(); }
  const float rsum = 1.0f / red[0];
  attn[row * C_CH + t] = e0 * rsum;
  if (has2) attn[row * C_CH + t + 256] = e1 * rsum;
}

// ---- T_hat = Vc @ sim^T, stored directly in KV_S layout [120*196, 4] ----
__global__ __launch_bounds__(32)
void that_gemm(const float* __restrict__ Vc, const float* __restrict__ sim,
               float* __restrict__ KVS)
{
  const int m0 = blockIdx.x * 16, n0 = blockIdx.y * 32;
  const int lane = threadIdx.x & 31, half = lane >> 4, l = lane & 15;
  const int m  = m0 + l;
  const int mA = m < N_TOK ? m : (N_TOK - 1);         // clamp, rows >=196 never stored
  const float* arow = Vc + mA * C_CH;
  const float* brow0 = sim + (n0 + l) * C_CH;         // B[k,n] = sim[n,k] (contiguous in k)
  const float* brow1 = sim + (n0 + 16 + l) * C_CH;
  v8f acc0 = {}, acc1 = {};
  for (int k0 = 0; k0 < C_CH; k0 += 4) {
    const int ka = k0 + 2 * half;
    const v2f a  = *(const v2f*)(arow + ka);
    const v2f b0 = *(const v2f*)(brow0 + ka);
    const v2f b1 = *(const v2f*)(brow1 + ka);
    acc0 = WMMA4(a, b0, acc0);
    acc1 = WMMA4(a, b1, acc1);
  }
  const int c0 = n0 + l, c1 = n0 + 16 + l;
  for (int i = 0; i < 8; ++i) {
    const int mm = m0 + half * 8 + i;                 // token index
    if (mm < N_TOK) {
      KVS[(((c0 >> 2) * N_TOK) + mm) * 4 + (c0 & 3)] = acc0[i];
      KVS[(((c1 >> 2) * N_TOK) + mm) * 4 + (c1 & 3)] = acc1[i];
    }
  }
}

// ---- per-head K/V (4x4 proj) + per-head K stats (shuffle + atomic) ----
__global__ __launch_bounds__(256)
void kv_heads(const float* __restrict__ KVS, const float* __restrict__ Wk,
              const float* __restrict__ Wv, float* __restrict__ Kh,
              float* __restrict__ Vh, float* __restrict__ stats)
{
  const int m = blockIdx.x * blockDim.x + threadIdx.x;
  const int lane = threadIdx.x & 31;
  const bool ok = m < M_KV;
  float x0 = 0, x1 = 0, x2 = 0, x3 = 0;
  if (ok) { x0 = KVS[m*4+0]; x1 = KVS[m*4+1]; x2 = KVS[m*4+2]; x3 = KVS[m*4+3]; }
  for (int h = 0; h < 4; ++h) {
    const float kv = x0*Wk[h] + x1*Wk[4+h] + x2*Wk[8+h] + x3*Wk[12+h];
    const float vv = x0*Wv[h] + x1*Wv[4+h] + x2*Wv[8+h] + x3*Wv[12+h];
    if (ok) { Kh[h*M_KV + m] = kv; Vh[h*M_KV + m] = vv; }
    float s = ok ? kv : 0.0f, s2 = ok ? kv*kv : 0.0f;
    for (int off = 16; off; off >>= 1) { s += __shfl_down(s, off); s2 += __shfl_down(s2, off); }
    if (lane == 0) { atomicAdd(&stats[2 + 2*h], s); atomicAdd(&stats[3 + 2*h], s2); }
  }
}

// ---- per-branch Q projection + per-(branch,head) Q stats ----
__global__ __launch_bounds__(224)
void q_proj(const float* __restrict__ e1, const float* __restrict__ e2,
            const float* __restrict__ e3, const float* __restrict__ e4,
            const float* __restrict__ W1, const float* __restrict__ W2,
            const float* __restrict__ W3, const float* __restrict__ W4,
            float* __restrict__ Qarr, float* __restrict__ stats)
{
  const int b = blockIdx.x;
  const float* emb = b == 0 ? e1 : (b == 1 ? e2 : (b == 2 ? e3 : e4));
  const float* Wq  = b == 0 ? W1 : (b == 1 ? W2 : (b == 2 ? W3 : W4));
  const int q = threadIdx.x, lane = threadIdx.x & 31;
  const bool ok = q < N_TOK;
  float x0 = 0, x1 = 0, x2 = 0, x3 = 0;
  if (ok) { x0 = emb[q*4+0]; x1 = emb[q*4+1]; x2 = emb[q*4+2]; x3 = emb[q*4+3]; }
  for (int h = 0; h < 4; ++h) {
    const float Qv = x0*Wq[h] + x1*Wq[4+h] + x2*Wq[8+h] + x3*Wq[12+h];
    if (ok) Qarr[(b*4 + h) * N_TOK + q] = Qv;
    float s = ok ? Qv : 0.0f, s2 = ok ? Qv*Qv : 0.0f;
    for (int off = 16; off; off >>= 1) { s += __shfl_down(s, off); s2 += __shfl_down(s2, off); }
    if (lane == 0) { atomicAdd(&stats[16 + 2*(b*4+h)], s); atomicAdd(&stats[17 + 2*(b*4+h)], s2); }
  }
}

// ---- combine outer-product stats: var(QK^T) = E[Q^2]E[K^2] - (mQ mK)^2 ----
__global__ void combine_stats(const float* __restrict__ stats,
                              const float* __restrict__ g2, float* __restrict__ scl)
{
  const int i = threadIdx.x;
  if (i < 16) {
    const int h = i & 3;
    const float mQ  = stats[16 + 2*i] / (float)N_TOK;
    const float EQ2 = stats[17 + 2*i] / (float)N_TOK;
    const float mK  = stats[2 + 2*h] / (float)M_KV;
    const float EK2 = stats[3 + 2*h] / (float)M_KV;
    const float m   = mQ * mK;
    const float var = EQ2 * EK2 - m * m;
    scl[i] = rsqrtf(var + I_EPS) * g2[h];     // softmax is shift-invariant: only scale matters
  }
}

// ---- branch attention: rank-1 softmax rows over 23520 keys, K in LDS ----
__global__ __launch_bounds__(1024)
void branch_attn(const float* __restrict__ Kh, const float* __restrict__ Vh,
                 const float* __restrict__ Qarr, const float* __restrict__ scl,
                 float* __restrict__ Carr)
{
  extern __shared__ float Ks[];                 // 23520 floats = 94 KB (of 320 KB WGP LDS)
  __shared__ float redA[32], redB[32], bc;
  const int b = blockIdx.z, h = blockIdx.y, qg = blockIdx.x;
  const int tid = threadIdx.x, lane = tid & 31, wid = tid >> 5;
  const float* Kg = Kh + h * M_KV;
  const float* Vg = Vh + h * M_KV;
  for (int i = tid; i < M_KV; i += 1024) Ks[i] = Kg[i];
  __syncthreads();
  const float scale = scl[b*4 + h];
  for (int qi = 0; qi < 28; ++qi) {             // 7 groups * 28 = 196
    const int q = qg * 28 + qi;
    const float alpha = Qarr[(b*4 + h) * N_TOK + q] * scale;
    float lm = -3.0e38f;
    for (int i = tid; i < M_KV; i += 1024) lm = fmaxf(lm, alpha * Ks[i]);
    for (int off = 16; off; off >>= 1) lm = fmaxf(lm, __shfl_down(lm, off));
    if (lane == 0) redA[wid] = lm;
    __syncthreads();
    if (wid == 0) {
      float v = redA[lane];
      for (int off = 16; off; off >>= 1) v = fmaxf(v, __shfl_down(v, off));
      if (lane == 0) bc = v;
    }
    __syncthreads();
    const float mx = bc;
    float se = 0.0f, sev = 0.0f;
    for (int i = tid; i < M_KV; i += 1024) {
      const float e = __expf(alpha * Ks[i] - mx);
      se += e; sev += e * Vg[i];
    }
    for (int off = 16; off; off >>= 1) { se += __shfl_down(se, off); sev += __shfl_down(sev, off); }
    if (lane == 0) { redA[wid] = se; redB[wid] = sev; }
    __syncthreads();
    if (wid == 0) {
      float a = redA[lane], v = redB[lane];
      for (int off = 16; off; off >>= 1) { a += __shfl_down(a, off); v += __shfl_down(v, off); }
      if (lane == 0) Carr[(b*4 + h) * N_TOK + q] = v / a;
    }
    __syncthreads();
  }
}

// ---- final per-branch output projection (4x4) ----
__global__ __launch_bounds__(256)
void out_proj(const float* __restrict__ Carr,
              const float* __restrict__ Wo1, const float* __restrict__ Wo2,
              const float* __restrict__ Wo3, const float* __restrict__ Wo4,
              float* __restrict__ out)
{
  const int idx = blockIdx.x * blockDim.x + threadIdx.x;   // b*196 + n
  if (idx >= 4 * N_TOK) return;
  const int b = idx / N_TOK, n = idx % N_TOK;
  const float* Wo = b == 0 ? Wo1 : (b == 1 ? Wo2 : (b == 2 ? Wo3 : Wo4));
  const float c0 = Carr[(b*4 + 0) * N_TOK + n];
  const float c1 = Carr[(b*4 + 1) * N_TOK + n];
  const float c2 = Carr[(b*4 + 2) * N_TOK + n];
  const float c3 = Carr[(b*4 + 3) * N_TOK + n];
  for (int e = 0; e < 4; ++e)
    out[idx * 4 + e] = c0*Wo[e] + c1*Wo[4+e] + c2*Wo[8+e] + c3*Wo[12+e];
}

// ---------------------------------------------------------------------------
extern "C" void kernel_launch(void* const* d_in, const int* in_sizes, int n_in,
                              void* d_out, int out_size, void* d_ws, size_t ws_size,
                              hipStream_t stream) {
  const float* emb1 = (const float*)d_in[0];
  const float* emb2 = (const float*)d_in[1];
  const float* emb3 = (const float*)d_in[2];
  const float* emb4 = (const float*)d_in[3];
  const float* embC = (const float*)d_in[4];
  const float* WqC  = (const float*)d_in[5];
  const float* WkC  = (const float*)d_in[6];
  const float* WvC  = (const float*)d_in[7];
  const float* Wq1  = (const float*)d_in[8];
  const float* Wq2  = (const float*)d_in[9];
  const float* Wq3  = (const float*)d_in[10];
  const float* Wq4  = (const float*)d_in[11];
  const float* Wk   = (const float*)d_in[12];
  const float* Wv   = (const float*)d_in[13];
  const float* Wo1  = (const float*)d_in[14];
  const float* Wo2  = (const float*)d_in[15];
  const float* Wo3  = (const float*)d_in[16];
  const float* Wo4  = (const float*)d_in[17];
  const float* g1   = (const float*)d_in[18];
  const float* b1   = (const float*)d_in[19];
  const float* g2   = (const float*)d_in[20];
  // const float* b2 = (const float*)d_in[21];  // shift-invariant under softmax

  float* ws    = (float*)d_ws;
  float* stats = ws + WS_STATS;
  float* scl   = ws + WS_SCL;
  float* Qc    = ws + WS_QC;
  float* Kc    = ws + WS_KC;
  float* Vc    = ws + WS_VC;
  float* attn  = ws + WS_ATTN;
  float* KVS   = ws + WS_KVS;
  float* Kh    = ws + WS_KH;
  float* Vh    = ws + WS_VH;
  float* Qarr  = ws + WS_QARR;
  float* Carr  = ws + WS_CARR;
  float* out   = (float*)d_out;

  init_stats<<<1, 64, 0, stream>>>(stats);
  qkv_gemm<<<dim3(13, 15, 3), 32, 0, stream>>>(embC, WqC, WkC, WvC, Qc, Kc, Vc);
  attn_gemm<<<dim3(30, 15), 32, 0, stream>>>(Qc, Kc, attn, stats);
  psi1_softmax<<<C_CH, 256, 0, stream>>>(attn, stats, g1, b1);
  that_gemm<<<dim3(13, 15), 32, 0, stream>>>(Vc, attn, KVS);
  kv_heads<<<(M_KV + 255) / 256, 256, 0, stream>>>(KVS, Wk, Wv, Kh, Vh, stats);
  q_proj<<<4, 224, 0, stream>>>(emb1, emb2, emb3, emb4, Wq1, Wq2, Wq3, Wq4, Qarr, stats);
  combine_stats<<<1, 16, 0, stream>>>(stats, g2, scl);
  branch_attn<<<dim3(7, 4, 4), 1024, M_KV * sizeof(float), stream>>>(Kh, Vh, Qarr, scl, Carr);
  out_proj<<<4, 256, 0, stream>>>(Carr, Wo1, Wo2, Wo3, Wo4, out);
}